// RdrSpcubeHead_35716948033609
// MI455X (gfx1250) — compile-verified
//
#include <hip/hip_runtime.h>
#include <math.h>

// Rotated-box IoU on gfx1250 (MI455X). Compute-bound elementwise geometry:
// no matmul structure -> no WMMA; CDNA5 async global->LDS staging +
// global_prefetch_b8 for the data path, minimized wave32 VALU for the math.

#ifndef __has_builtin
#define __has_builtin(x) 0
#endif

#if defined(__HIP_DEVICE_COMPILE__) && __has_builtin(__builtin_amdgcn_global_load_async_to_lds_b32)
#define RBIOU_ASYNC 1
#else
#define RBIOU_ASYNC 0
#endif

#define TPB 256
#define RB_EPS 1e-8f
#define VROW 52   // padded per-thread LDS row (dwords): 24 float2 + 4 pad, 2-way conflicts only

__device__ __forceinline__ float rb_rcp(float x) { return __builtin_amdgcn_rcpf(x); }

__device__ __forceinline__ void rbiou_wait_async() {
#if RBIOU_ASYNC
#if __has_builtin(__builtin_amdgcn_s_wait_asynccnt)
  __builtin_amdgcn_s_wait_asynccnt(0);
#else
  asm volatile("s_wait_asynccnt 0" ::: "memory");
#endif
#endif
}

#if RBIOU_ASYNC
typedef __attribute__((address_space(1))) void as1_void;
typedef __attribute__((address_space(3))) void as3_void;
typedef __attribute__((address_space(1))) int  as1_int;
typedef __attribute__((address_space(3))) int  as3_int;
__device__ __forceinline__ void rbiou_async_dword(const float* g, float* l) {
  __builtin_amdgcn_global_load_async_to_lds_b32(
      (as1_int*)(as1_void*)(void*)g,
      (as3_int*)(as3_void*)(void*)l,
      /*offset=*/0, /*cpol=*/0);
}
#endif

__global__ __launch_bounds__(TPB) void rbiou_kernel(const float* __restrict__ box1,
                                                    const float* __restrict__ box2,
                                                    float* __restrict__ out,
                                                    long long P) {
  __shared__ float s1[TPB * 5];
  __shared__ float s2[TPB * 5];
  __shared__ float vtab[TPB * VROW];
  const int tid = threadIdx.x;
  const long long blockBase = (long long)blockIdx.x * TPB;
  const long long eBase = blockBase * 5;
  const long long totE = P * 5;

  // ---- stage this block's AoS tile into LDS, fully coalesced dwords ----
#pragma unroll
  for (int d = 0; d < 5; ++d) {
    int idx = tid + d * TPB;
    long long g = eBase + idx;
    if (g >= totE) g = totE - 1;
#if RBIOU_ASYNC
    rbiou_async_dword(box1 + g, s1 + idx);
    rbiou_async_dword(box2 + g, s2 + idx);
#else
    s1[idx] = box1[g];
    s2[idx] = box2[g];
#endif
  }
  // prefetch next block's tiles (speculative, gfx1250 global_prefetch_b8)
  {
    long long pf = eBase + (long long)TPB * 5;
    if (pf < totE) {
      __builtin_prefetch(box1 + pf, 0, 0);
      __builtin_prefetch(box2 + pf, 0, 0);
    }
  }
  rbiou_wait_async();
  __syncthreads();

  const long long i = blockBase + tid;
  if (i >= P) return;

  float b1[5], b2[5];
#pragma unroll
  for (int d = 0; d < 5; ++d) { b1[d] = s1[tid * 5 + d]; b2[d] = s2[tid * 5 + d]; }

  // ---- corners: angles are in [-pi,pi] by construction, so HW v_sin/v_cos
  //      (no range reduction needed) are safe and ~2 ops each ----
  float c1x[4], c1y[4], c2x[4], c2y[4];
  {
    const float hx[4] = {0.5f, -0.5f, -0.5f, 0.5f};
    const float hy[4] = {0.5f, 0.5f, -0.5f, -0.5f};
    float sn = __sinf(b1[4]), cs = __cosf(b1[4]);
#pragma unroll
    for (int k = 0; k < 4; ++k) {
      float px = hx[k] * b1[2], py = hy[k] * b1[3];
      c1x[k] = px * cs - py * sn + b1[0];
      c1y[k] = px * sn + py * cs + b1[1];
    }
    sn = __sinf(b2[4]); cs = __cosf(b2[4]);
#pragma unroll
    for (int k = 0; k < 4; ++k) {
      float px = hx[k] * b2[2], py = hy[k] * b2[3];
      c2x[k] = px * cs - py * sn + b2[0];
      c2y[k] = px * sn + py * cs + b2[1];
    }
  }

  bool  m[24];
  float vx[24], vy[24];
#pragma unroll
  for (int k = 0; k < 4; ++k) {
    vx[k] = c1x[k]; vy[k] = c1y[k];
    vx[4 + k] = c2x[k]; vy[4 + k] = c2y[k];
  }

  // ---- corners of box1 inside box2 (m[0..3]) ----
  {
    float ax = c2x[0], ay = c2y[0];
    float abx = c2x[1] - ax, aby = c2y[1] - ay;
    float adx = c2x[3] - ax, ady = c2y[3] - ay;
    float rnab = rb_rcp(abx * abx + aby * aby);
    float rnad = rb_rcp(adx * adx + ady * ady);
#pragma unroll
    for (int k = 0; k < 4; ++k) {
      float amx = c1x[k] - ax, amy = c1y[k] - ay;
      float rab = (abx * amx + aby * amy) * rnab;
      float rad = (adx * amx + ady * amy) * rnad;
      m[k] = (rab > -1e-6f) && (rab < 1.000001f) && (rad > -1e-6f) && (rad < 1.000001f);
    }
  }
  // ---- corners of box2 inside box1 (m[4..7]) ----
  {
    float ax = c1x[0], ay = c1y[0];
    float abx = c1x[1] - ax, aby = c1y[1] - ay;
    float adx = c1x[3] - ax, ady = c1y[3] - ay;
    float rnab = rb_rcp(abx * abx + aby * aby);
    float rnad = rb_rcp(adx * adx + ady * ady);
#pragma unroll
    for (int k = 0; k < 4; ++k) {
      float amx = c2x[k] - ax, amy = c2y[k] - ay;
      float rab = (abx * amx + aby * amy) * rnab;
      float rad = (adx * amx + ady * amy) * rnad;
      m[4 + k] = (rab > -1e-6f) && (rab < 1.000001f) && (rad > -1e-6f) && (rad < 1.000001f);
    }
  }

  // ---- 16 edge-edge intersections (m[8..23], vertices 8..23) ----
#pragma unroll
  for (int e1 = 0; e1 < 4; ++e1) {
    float x1 = c1x[e1], y1 = c1y[e1];
    float x2 = c1x[(e1 + 1) & 3], y2 = c1y[(e1 + 1) & 3];
#pragma unroll
    for (int e2 = 0; e2 < 4; ++e2) {
      float x3 = c2x[e2], y3 = c2y[e2];
      float x4 = c2x[(e2 + 1) & 3], y4 = c2y[(e2 + 1) & 3];
      float num  = (x1 - x2) * (y3 - y4) - (y1 - y2) * (x3 - x4);
      float dent = (x1 - x3) * (y3 - y4) - (y1 - y3) * (x3 - x4);
      float denu = (x1 - x2) * (y1 - y3) - (y1 - y2) * (x1 - x3);
      bool  zero = (num == 0.0f);
      float rn = rb_rcp(zero ? 1.0f : num);
      float t = zero ? -1.0f : (dent * rn);
      float u = zero ? -1.0f : (-denu * rn);
      bool  mk = (t > 0.f) && (t < 1.f) && (u > 0.f) && (u < 1.f);
      float t2 = dent * rb_rcp(num + RB_EPS);
      int idx = 8 + e1 * 4 + e2;
      vx[idx] = mk ? (x1 + t2 * (x2 - x1)) : 0.f;
      vy[idx] = mk ? (y1 + t2 * (y2 - y1)) : 0.f;
      m[idx] = mk;
    }
  }

  // ---- spill vertex table to a private LDS row (enables 1-op dynamic gather) ----
  float* row = vtab + tid * VROW;
#pragma unroll
  for (int j = 0; j < 24; j += 2) {
    *(float4*)(row + 2 * j) = make_float4(vx[j], vy[j], vx[j + 1], vy[j + 1]);
  }

  // ---- masked centroid + valid count ----
  int nv = 0; float sxm = 0.f, sym = 0.f;
#pragma unroll
  for (int k = 0; k < 24; ++k) {
    nv  += m[k] ? 1 : 0;
    sxm += m[k] ? vx[k] : 0.f;
    sym += m[k] ? vy[k] : 0.f;
  }
  float rdenom = rb_rcp((float)(nv > 1 ? nv : 1));
  float mxv = sxm * rdenom, myv = sym * rdenom;

  // ---- sortable-uint keys: pseudo-angle p in [-2,2] (monotone with atan2);
  //      p+2 >= 0 so its raw IEEE bits are already order-preserving; pack the
  //      candidate index into the low 5 bits (stable ties, unique keys) ----
  unsigned comb[24];
#pragma unroll
  for (int k = 0; k < 24; ++k) {
    float dx = vx[k] - mxv, dy = vy[k] - myv;
    float mag = fabsf(dx) + fabsf(dy);
    float t = (mag > 0.f) ? (dx * rb_rcp(mag)) : 1.0f;  // (0,0) -> key 0, like atan2(0,0)
    float p = copysignf(1.0f - t, dy);                  // monotone with atan2 over (-pi,pi]
    unsigned s = __float_as_uint(p + 2.0f);             // non-negative -> bit-order == value-order
    s = (s & 0xFFFFFFE0u) | (unsigned)k;
    comb[k] = m[k] ? s : (0xFFFFFF00u | (unsigned)k);   // masked -> above all valid keys
  }

  // ---- 8 smallest, sorted, via online insertion (pure v_min_u32/v_max_u32,
  //      VOPD dual-issue friendly); keys unique so order is exact ----
  unsigned r[8];
#pragma unroll
  for (int k = 0; k < 8; ++k) r[k] = 0xFFFFFFFFu;
#pragma unroll
  for (int j = 0; j < 24; ++j) {
    unsigned c = comb[j];
#pragma unroll
    for (int k = 0; k < 8; ++k) {
      unsigned lo = (c < r[k]) ? c : r[k];   // v_min_u32
      unsigned hi = (c < r[k]) ? r[k] : c;   // v_max_u32
      r[k] = lo;
      c = hi;
    }
  }

  // ---- gather the selected vertices (one ds_load_b64 each) ----
  float sx[8], sy[8];
#pragma unroll
  for (int k = 0; k < 8; ++k) {
    int bi = (int)(r[k] & 31u);              // always <24: 24 candidates >= 8 slots
    float2 gv = *(const float2*)(row + 2 * bi);
    bool valid = (k < nv);
    sx[k] = valid ? gv.x : 0.f;
    sy[k] = valid ? gv.y : 0.f;
  }

  // ---- shoelace over sel9 (close polygon by duplicating first at slot nv) ----
  float total = 0.f;
#pragma unroll
  for (int k = 0; k < 8; ++k) {
    float ax = (k == nv) ? sx[0] : sx[k];
    float ay = (k == nv) ? sy[0] : sy[k];
    float bx, by;
    if (k == 7) { bx = (8 == nv) ? sx[0] : 0.f; by = (8 == nv) ? sy[0] : 0.f; }
    else        { bx = (k + 1 == nv) ? sx[0] : sx[k + 1];
                  by = (k + 1 == nv) ? sy[0] : sy[k + 1]; }
    total += ax * by - ay * bx;
  }
  float area = fabsf(total) * 0.5f;

  // ---- IoU and outputs: [iou | corners1 | corners2 | u] ----
  float area1 = b1[2] * b1[3], area2 = b2[2] * b2[3];
  float uVal = area1 + area2 - area;
  out[i] = area / uVal;                    // exact divide: this is an output value
  out[P * 17 + i] = uVal;

  float4* oc1 = (float4*)(out + P) + i * 2;
  oc1[0] = make_float4(c1x[0], c1y[0], c1x[1], c1y[1]);
  oc1[1] = make_float4(c1x[2], c1y[2], c1x[3], c1y[3]);
  float4* oc2 = (float4*)(out + P * 9) + i * 2;
  oc2[0] = make_float4(c2x[0], c2y[0], c2x[1], c2y[1]);
  oc2[1] = make_float4(c2x[2], c2y[2], c2x[3], c2y[3]);
}

extern "C" void kernel_launch(void* const* d_in, const int* in_sizes, int n_in,
                              void* d_out, int out_size, void* d_ws, size_t ws_size,
                              hipStream_t stream) {
  const float* box1 = (const float*)d_in[0];
  const float* box2 = (const float*)d_in[1];
  float* out = (float*)d_out;
  long long P = (long long)in_sizes[0] / 5;   // B*N box pairs
  int blocks = (int)((P + TPB - 1) / TPB);
  hipLaunchKernelGGL(rbiou_kernel, dim3(blocks), dim3(TPB), 0, stream,
                     box1, box2, out, P);
}